// MixedEmbeddingEncoder_62173946577609
// MI455X (gfx1250) — compile-verified
//
#include <hip/hip_runtime.h>

// Problem dims (fixed by the reference harness)
constexpr int Bq = 32;
constexpr int Sq = 4096;
constexpr int Dq = 768;
constexpr int Wq = 2048;
constexpr int NSEG = Bq * Wq + 1;          // segment bins (+1 dummy, unused)
constexpr int ROWS = Bq * (Sq - 1);        // 131040 valid rows (s >= 1)

typedef float v2f __attribute__((ext_vector_type(2)));
typedef float v4f __attribute__((ext_vector_type(4)));
typedef float v8f __attribute__((ext_vector_type(8)));

// ---------------------------------------------------------------------------
// Kernel 1: zero the segment-sum / segment-count bins in workspace.
// ---------------------------------------------------------------------------
__global__ void zero_bins_kernel(float* __restrict__ bins, int n) {
    int i = blockIdx.x * blockDim.x + threadIdx.x;
    if (i < n) bins[i] = 0.0f;
}

// ---------------------------------------------------------------------------
// Exact f32 cross-lane (wave32) sum using the matrix unit:
//   A (16x4 f32): lane partials at K=0 (lanes 0-15) and K=2 (lanes 16-31),
//   B (4x16 f32): all ones  ->  D[m][n] = p_m + p_{m+16}.
//   Summing the 8 D VGPRs per lane gives one half; shfl_xor(16) combines.
// ---------------------------------------------------------------------------
__device__ inline float wave_sum32_wmma(float p) {
    v2f a; a[0] = p;    a[1] = 0.0f;
    v2f b; b[0] = 1.0f; b[1] = 1.0f;
    v8f c = {};
    c = __builtin_amdgcn_wmma_f32_16x16x4_f32(
            /*neg_a=*/false, a, /*neg_b=*/false, b,
            /*c_mod=*/(short)0, c, /*reuse_a=*/false, /*reuse_b=*/false);
    float h = ((c[0] + c[1]) + (c[2] + c[3])) + ((c[4] + c[5]) + (c[6] + c[7]));
    h += __shfl_xor(h, 16, 32);   // combine the two 8-row halves
    return h;
}

// ---------------------------------------------------------------------------
// Kernel 2: one wave32 per row (b, s>=1). Coalesced NT b128 loads of the
// 768-float row, per-lane partial (24 elems), WMMA wave reduction, then
// lane 0 atomically accumulates into the (b*W + word) bin.
// ---------------------------------------------------------------------------
__global__ __launch_bounds__(256) void reduce_rows_kernel(
    const float* __restrict__ emb,
    const int* __restrict__ orig,
    const unsigned char* __restrict__ mask,
    const int* __restrict__ num_words_p,
    float* __restrict__ sums,
    float* __restrict__ cnts)
{
    const int nw   = num_words_p[0];
    const int wid  = (blockIdx.x * 256 + threadIdx.x) >> 5;   // global wave id == row id
    const int lane = threadIdx.x & 31;
    const int b = wid / (Sq - 1);
    const int s = 1 + wid % (Sq - 1);

    const v4f* p = (const v4f*)(emb + ((size_t)b * Sq + s) * Dq) + lane;
    v4f acc = __builtin_nontemporal_load(p);            // streamed once: NT hint
#pragma unroll
    for (int i = 1; i < 6; ++i) {
        v4f v = __builtin_nontemporal_load(p + i * 32); // 32 lanes * 16B = 512B/iter
        acc += v;
    }
    float partial = (acc[0] + acc[1]) + (acc[2] + acc[3]);

    float total = wave_sum32_wmma(partial);             // exact f32 row sum

    if (lane == 0) {
        const size_t idx = (size_t)b * Sq + s;
        if (mask[idx]) {
            const int seg = orig[idx] + b * nw;
            atomicAdd(&sums[seg], total);
            atomicAdd(&cnts[seg], 1.0f);
        }
    }
}

// ---------------------------------------------------------------------------
// Kernel 3: one block per (b, w). Compute scal = sums/(cnts*D) (0 if empty)
// and broadcast it across D with NT b128 stores.
// ---------------------------------------------------------------------------
__global__ __launch_bounds__(192) void broadcast_out_kernel(
    const float* __restrict__ sums,
    const float* __restrict__ cnts,
    float* __restrict__ out)
{
    const int seg = blockIdx.x;                 // == b*W + w
    const float cn = cnts[seg];
    const float sm = sums[seg];
    const float scal = (cn > 0.0f) ? sm / (cn * (float)Dq) : 0.0f;

    v4f v; v[0] = scal; v[1] = scal; v[2] = scal; v[3] = scal;
    v4f* o = (v4f*)(out + (size_t)seg * Dq) + threadIdx.x;   // 192 threads * 4 = 768
    __builtin_nontemporal_store(v, o);          // 201MB stream: keep out of caches
}

// ---------------------------------------------------------------------------
extern "C" void kernel_launch(void* const* d_in, const int* in_sizes, int n_in,
                              void* d_out, int out_size, void* d_ws, size_t ws_size,
                              hipStream_t stream)
{
    const float*          emb  = (const float*)d_in[0];
    const int*            orig = (const int*)d_in[1];
    const unsigned char*  mask = (const unsigned char*)d_in[2];
    const int*            nw   = (const int*)d_in[3];
    float*                out  = (float*)d_out;

    float* sums = (float*)d_ws;         // NSEG floats
    float* cnts = sums + NSEG;          // NSEG floats

    const int nbins = 2 * NSEG;
    zero_bins_kernel<<<(nbins + 255) / 256, 256, 0, stream>>>(sums, nbins);

    // ROWS = 131040 rows, 8 waves/block -> exactly 16380 full blocks (EXEC all-1s)
    reduce_rows_kernel<<<ROWS / 8, 256, 0, stream>>>(emb, orig, mask, nw, sums, cnts);

    broadcast_out_kernel<<<Bq * Wq, 192, 0, stream>>>(sums, cnts, out);
}